// FactorizedDepthwise3DConv_78967268704390
// MI455X (gfx1250) — compile-verified
//
#include <hip/hip_runtime.h>
#include <hip/hip_bf16.h>

// Problem constants (from reference setup_inputs)
#define BB   8
#define CC   64
#define TT   16
#define HH   56
#define WW   56
#define HWSZ (HH * WW)          // 3136
#define THW  (TT * HWSZ)        // 50176
#define COUT 128
#define NTILES (THW / 16)       // 3136
#define MTILES (COUT / 16)      // 8
#define GN_GROUPS 4
#define GN_EPS 1e-5f

#define HALO 58                 // 56 + 2 halo

typedef float v2f __attribute__((ext_vector_type(2)));
typedef float v8f __attribute__((ext_vector_type(8)));

// ---------------------------------------------------------------------------
// K1: fused depthwise spatial 3x3 conv + depthwise temporal 3-tap conv.
// One block per (b, c, t). Stages 3 haloed 58x58 input planes (t-1,t,t+1) in
// LDS, then each thread computes z = sum_p tw[p]*(conv3x3(x[t-1+p]) + sb) + tb.
// ---------------------------------------------------------------------------
__global__ __launch_bounds__(256)
void k1_dwconv(const float* __restrict__ x,
               const float* __restrict__ sw,   // (C,1,3,3)
               const float* __restrict__ sb,   // (C)
               const float* __restrict__ tw,   // (C,1,3,1,1)
               const float* __restrict__ tb,   // (C)
               float* __restrict__ z)          // (B,C,T,H,W)
{
    __shared__ float xs[3][HALO][HALO];        // 40368 bytes

    const int blk = blockIdx.x;                // b*C*T + c*T + t
    const int t = blk % TT;
    const int c = (blk / TT) % CC;
    const int b = blk / (TT * CC);
    const int tid = threadIdx.x;

    const float* __restrict__ xc = x + ((size_t)(b * CC + c)) * THW;

    // Cooperative haloed load of 3 temporal planes (zero padding).
    for (int i = tid; i < 3 * HALO * HALO; i += 256) {
        const int p  = i / (HALO * HALO);
        const int r  = i % (HALO * HALO);
        const int hh = r / HALO - 1;
        const int ww = r % HALO - 1;
        const int tt = t - 1 + p;
        float v = 0.0f;
        if (tt >= 0 && tt < TT && hh >= 0 && hh < HH && ww >= 0 && ww < WW)
            v = xc[(size_t)tt * HWSZ + hh * WW + ww];
        (&xs[0][0][0])[i] = v;
    }
    __syncthreads();

    const float w00 = sw[c * 9 + 0], w01 = sw[c * 9 + 1], w02 = sw[c * 9 + 2];
    const float w10 = sw[c * 9 + 3], w11 = sw[c * 9 + 4], w12 = sw[c * 9 + 5];
    const float w20 = sw[c * 9 + 6], w21 = sw[c * 9 + 7], w22 = sw[c * 9 + 8];
    const float bias_s = sb[c];
    const float twk0 = tw[c * 3 + 0], twk1 = tw[c * 3 + 1], twk2 = tw[c * 3 + 2];
    const float bias_t = tb[c];

    float* __restrict__ zc = z + (((size_t)(b * CC + c)) * TT + t) * HWSZ;

    for (int pos = tid; pos < HWSZ; pos += 256) {
        const int h = pos / WW;
        const int w = pos % WW;
        float acc = bias_t;
        const float tk[3] = { twk0, twk1, twk2 };
#pragma unroll
        for (int p = 0; p < 3; ++p) {
            float y = xs[p][h + 0][w + 0] * w00 + xs[p][h + 0][w + 1] * w01 +
                      xs[p][h + 0][w + 2] * w02 +
                      xs[p][h + 1][w + 0] * w10 + xs[p][h + 1][w + 1] * w11 +
                      xs[p][h + 1][w + 2] * w12 +
                      xs[p][h + 2][w + 0] * w20 + xs[p][h + 2][w + 1] * w21 +
                      xs[p][h + 2][w + 2] * w22;
            y += bias_s;
            acc += tk[p] * y;
        }
        zc[pos] = acc;
    }
}

// ---------------------------------------------------------------------------
// K2: pointwise 1x1x1 conv (channel GEMM, 64 -> 128) via V_WMMA_F32_16X16X4_F32.
// One wave per 16x16 (Cout x positions) tile; K=64 in 16 steps of k=4.
// Writes p (+bias) to out and per-wave (sum, sumsq) partials for GroupNorm.
// ---------------------------------------------------------------------------
__global__ __launch_bounds__(256)
void k2_pointwise_wmma(const float* __restrict__ z,     // (B,C,THW)
                       const float* __restrict__ pw,    // (COUT,C) row-major
                       const float* __restrict__ pb,    // (COUT)
                       float* __restrict__ out,         // (B,COUT,THW)
                       float* __restrict__ partial)     // (waves, 2)
{
    const int gtid = blockIdx.x * 256 + threadIdx.x;
    const int wave = gtid >> 5;
    const int lane = gtid & 31;

    const int nt = wave % NTILES;
    const int mt = (wave / NTILES) % MTILES;
    const int b  = wave / (NTILES * MTILES);

    const int l  = lane & 15;       // column within half-wave
    const int hi = lane >> 4;       // 0: lanes 0-15, 1: lanes 16-31
    const int m0 = mt * 16;
    const int n0 = nt * 16;

    const float* __restrict__ zb = z + (size_t)b * CC * THW;

    v8f acc = {};
#pragma unroll
    for (int k = 0; k < CC; k += 4) {
        const int ka = k + hi * 2;
        // A (16x4 fp32): V0 = A[M=l][K=ka], V1 = A[M=l][K=ka+1]
        v2f a;
        a.x = pw[(m0 + l) * CC + ka];
        a.y = pw[(m0 + l) * CC + ka + 1];
        // B (4x16 fp32): V0 = B[K=ka][N=l], V1 = B[K=ka+1][N=l]
        const float* __restrict__ zp = zb + (size_t)ka * THW + n0 + l;
        v2f bm;
        bm.x = zp[0];
        bm.y = zp[THW];
        acc = __builtin_amdgcn_wmma_f32_16x16x4_f32(
            /*neg_a=*/false, a, /*neg_b=*/false, bm,
            /*c_mod=*/(short)0, acc, /*reuse_a=*/false, /*reuse_b=*/false);
    }

    // Store p + bias, accumulate group-norm partial sums.
    float s1 = 0.0f, s2 = 0.0f;
    float* __restrict__ ob = out + (size_t)b * COUT * THW;
#pragma unroll
    for (int j = 0; j < 8; ++j) {
        const int m = m0 + j + hi * 8;     // C/D layout: VGPR j -> rows j / j+8
        const float v = acc[j] + pb[m];
        ob[(size_t)m * THW + n0 + l] = v;
        s1 += v;
        s2 += v * v;
    }

    // Deterministic in-wave reduction (wave32).
    for (int off = 16; off > 0; off >>= 1) {
        s1 += __shfl_down(s1, off, 32);
        s2 += __shfl_down(s2, off, 32);
    }
    if (lane == 0) {
        partial[(size_t)wave * 2 + 0] = s1;
        partial[(size_t)wave * 2 + 1] = s2;
    }
}

// ---------------------------------------------------------------------------
// K2b: deterministic reduction of per-wave partials into per-(b,group) stats.
// The 6272 partials of one (b, group) are contiguous: mt in {2g, 2g+1}.
// ---------------------------------------------------------------------------
__global__ __launch_bounds__(256)
void k2b_reduce(const float* __restrict__ partial,
                float* __restrict__ stats)     // (B*4, 2)
{
    __shared__ float r1[256];
    __shared__ float r2[256];
    const int bg = blockIdx.x;                 // b*4 + g
    const int b = bg >> 2;
    const int g = bg & 3;
    const int base = (b * MTILES + g * 2) * NTILES;   // contiguous 2*NTILES waves
    const int count = 2 * NTILES;                     // 6272

    float s1 = 0.0f, s2 = 0.0f;
    for (int i = threadIdx.x; i < count; i += 256) {
        s1 += partial[(size_t)(base + i) * 2 + 0];
        s2 += partial[(size_t)(base + i) * 2 + 1];
    }
    r1[threadIdx.x] = s1;
    r2[threadIdx.x] = s2;
    __syncthreads();
    for (int s = 128; s > 0; s >>= 1) {
        if (threadIdx.x < s) {
            r1[threadIdx.x] += r1[threadIdx.x + s];
            r2[threadIdx.x] += r2[threadIdx.x + s];
        }
        __syncthreads();
    }
    if (threadIdx.x == 0) {
        stats[bg * 2 + 0] = r1[0];
        stats[bg * 2 + 1] = r2[0];
    }
}

// ---------------------------------------------------------------------------
// K3: GroupNorm + ReLU in place on out, float4 vectorized.
// ---------------------------------------------------------------------------
__global__ __launch_bounds__(256)
void k3_gn_relu(float* __restrict__ out,
                const float* __restrict__ stats,
                const float* __restrict__ gw,
                const float* __restrict__ gb)
{
    const float inv_cnt = 1.0f / ((float)(COUT / GN_GROUPS) * (float)THW);
    const size_t total4 = (size_t)BB * COUT * THW / 4;
    for (size_t i = (size_t)blockIdx.x * 256 + threadIdx.x; i < total4;
         i += (size_t)gridDim.x * 256) {
        const size_t e = i * 4;                    // THW divisible by 4 -> same o
        const int o = (int)((e / THW) % COUT);
        const int b = (int)(e / ((size_t)COUT * THW));
        const int g = o >> 5;
        const float s1 = stats[(b * GN_GROUPS + g) * 2 + 0];
        const float s2 = stats[(b * GN_GROUPS + g) * 2 + 1];
        const float mean = s1 * inv_cnt;
        const float var  = s2 * inv_cnt - mean * mean;
        const float inv  = rsqrtf(var + GN_EPS);
        const float scale = inv * gw[o];
        const float shift = gb[o] - mean * scale;

        float4 v = reinterpret_cast<float4*>(out)[i];
        v.x = fmaxf(v.x * scale + shift, 0.0f);
        v.y = fmaxf(v.y * scale + shift, 0.0f);
        v.z = fmaxf(v.z * scale + shift, 0.0f);
        v.w = fmaxf(v.w * scale + shift, 0.0f);
        reinterpret_cast<float4*>(out)[i] = v;
    }
}

// ---------------------------------------------------------------------------
// Launch
// ---------------------------------------------------------------------------
extern "C" void kernel_launch(void* const* d_in, const int* in_sizes, int n_in,
                              void* d_out, int out_size, void* d_ws, size_t ws_size,
                              hipStream_t stream) {
    const float* x  = (const float*)d_in[0];
    const float* sw = (const float*)d_in[1];
    const float* sb = (const float*)d_in[2];
    const float* tw = (const float*)d_in[3];
    const float* tb = (const float*)d_in[4];
    const float* pw = (const float*)d_in[5];
    const float* pb = (const float*)d_in[6];
    const float* gw = (const float*)d_in[7];
    const float* gb = (const float*)d_in[8];
    float* out = (float*)d_out;

    // Workspace layout: z | per-wave partials | stats
    const size_t z_elems = (size_t)BB * CC * THW;          // 25,690,112 floats
    const size_t n_waves = (size_t)BB * MTILES * NTILES;   // 200,704
    float* z       = (float*)d_ws;
    float* partial = z + z_elems;
    float* stats   = partial + n_waves * 2;

    // K1: depthwise spatial+temporal conv -> z
    k1_dwconv<<<BB * CC * TT, 256, 0, stream>>>(x, sw, sb, tw, tb, z);

    // K2: WMMA pointwise GEMM -> out (pre-norm) + partial stats
    const int k2_blocks = (int)(n_waves / 8);              // 8 waves per 256-thr block
    k2_pointwise_wmma<<<k2_blocks, 256, 0, stream>>>(z, pw, pb, out, partial);

    // K2b: reduce partials -> per-(b,group) stats
    k2b_reduce<<<BB * GN_GROUPS, 256, 0, stream>>>(partial, stats);

    // K3: GroupNorm + ReLU in place
    k3_gn_relu<<<4096, 256, 0, stream>>>(out, stats, gw, gb);
}